// KbNuFFT2d_torch_71897752535094
// MI455X (gfx1250) — compile-verified
//
#include <hip/hip_runtime.h>
#include <hip/hip_bf16.h>

// ---------------------------------------------------------------------------
// KB-NuFFT forward on gfx1250 (MI455X).
// FFT realized as DFT-matrix GEMMs on the fp32 WMMA pipe
// (V_WMMA_F32_16X16X4_F32) for full-precision matrix math; gather stage is
// L2-resident (grids = 16 MB << 192 MB L2).
// ---------------------------------------------------------------------------

#define HH   256
#define WW   256
#define GG   512          // oversampled grid (2x)
#define BB   8
#define JJ   7

typedef float v2f __attribute__((ext_vector_type(2)));
typedef float v8f __attribute__((ext_vector_type(8)));

// ---------------- DFT matrix generation: F[k][n] = e^{-2pi i kn/512}/sqrt(512)
__global__ void fgen_kernel(float* __restrict__ Fc, float* __restrict__ Fs) {
    int idx = blockIdx.x * blockDim.x + threadIdx.x;
    if (idx >= GG * GG) return;
    int k = idx >> 9;          // /512
    int n = idx & (GG - 1);
    int p = (k * n) & (GG - 1);                 // exact mod keeps angle precise
    float theta = (float)p * (6.283185307179586f / (float)GG);
    float s, c;
    sincosf(theta, &s, &c);
    const float norm = 0.04419417382415922f;    // 1/sqrt(512); applied twice -> 1/512 ortho
    Fc[idx] = c * norm;
    Fs[idx] = -s * norm;
}

// ---------------- apodization: X = (xr + i xi) * sc_y[y] * sc_x[x]
__global__ void apod_kernel(const float* __restrict__ xr, const float* __restrict__ xi,
                            const float* __restrict__ sc_y, const float* __restrict__ sc_x,
                            float* __restrict__ Xr, float* __restrict__ Xi) {
    int idx = blockIdx.x * blockDim.x + threadIdx.x;
    if (idx >= BB * HH * WW) return;
    int r = idx & (HH * WW - 1);
    int y = r >> 8;            // /256
    int x = r & (WW - 1);
    float s = sc_y[y] * sc_x[x];
    Xr[idx] = xr[idx] * s;
    Xi[idx] = xi[idx] * s;
}

// ---------------- WMMA fragment helpers (ISA 7.12.2 fp32 layouts) -----------
__device__ __forceinline__ v2f load_a_frag(const float* __restrict__ A, int lda,
                                           int m, int k, int lane) {
    // A 16x4: lanes 0-15 -> K={0,1}, lanes 16-31 -> K={2,3}; M = lane%16
    int mm = m + (lane & 15);
    int kk = k + ((lane >> 4) << 1);
    const float* p = A + (size_t)mm * lda + kk;
    v2f r; r.x = p[0]; r.y = p[1];
    return r;
}

__device__ __forceinline__ v2f load_b_frag(const float* __restrict__ B, int ldb,
                                           int k, int n, int lane) {
    // B 4x16: N = lane%16; lanes 0-15 -> K={0,1}, lanes 16-31 -> K={2,3}
    int nn = n + (lane & 15);
    int kk = k + ((lane >> 4) << 1);
    v2f r;
    r.x = B[(size_t)kk * ldb + nn];
    r.y = B[(size_t)(kk + 1) * ldb + nn];
    return r;
}

__device__ __forceinline__ void store_c_frag(float* __restrict__ C, int ldc,
                                             int m, int n, int lane, v8f c) {
    // C/D 16x16: VGPR v holds M = v + 8*(lane/16); N = lane%16
    int nn = n + (lane & 15);
    int mb = m + ((lane >> 4) << 3);
#pragma unroll
    for (int v = 0; v < 8; ++v)
        C[(size_t)(mb + v) * ldc + nn] = c[v];
}

// ---------------- complex GEMM via fp32 WMMA --------------------------------
// C = A * B (complex, split planes). Block = 256 threads = 8 waves, wave tile
// 32x32 (2x2 WMMA tiles), workgroup tile 64(M) x 128(N). All dims divide
// evenly -> EXEC stays all-ones around every WMMA (ISA requirement).
__global__ void __launch_bounds__(256)
cgemm_wmma_kernel(const float* __restrict__ Ar, const float* __restrict__ Ai,
                  int lda, long strideA,
                  const float* __restrict__ Br, const float* __restrict__ Bi,
                  int ldb, long strideB,
                  float* __restrict__ Cr, float* __restrict__ Ci,
                  int ldc, long strideC, int K) {
    const int lane = threadIdx.x & 31;
    const int wid  = threadIdx.x >> 5;
    const int b    = blockIdx.z;

    Ar += (size_t)b * strideA;  Ai += (size_t)b * strideA;
    Br += (size_t)b * strideB;  Bi += (size_t)b * strideB;
    Cr += (size_t)b * strideC;  Ci += (size_t)b * strideC;

    const int mBase = blockIdx.x * 64  + (wid >> 2) * 32;
    const int nBase = blockIdx.y * 128 + (wid & 3)  * 32;

    v8f aRR[2][2], aII[2][2], aRI[2][2], aIR[2][2];
#pragma unroll
    for (int i = 0; i < 2; ++i)
#pragma unroll
        for (int j = 0; j < 2; ++j) {
            aRR[i][j] = (v8f)0.0f; aII[i][j] = (v8f)0.0f;
            aRI[i][j] = (v8f)0.0f; aIR[i][j] = (v8f)0.0f;
        }

    for (int k = 0; k < K; k += 4) {
        v2f ar[2], ai[2], br[2], bi[2];
#pragma unroll
        for (int i = 0; i < 2; ++i) {
            ar[i] = load_a_frag(Ar, lda, mBase + 16 * i, k, lane);
            ai[i] = load_a_frag(Ai, lda, mBase + 16 * i, k, lane);
        }
#pragma unroll
        for (int j = 0; j < 2; ++j) {
            br[j] = load_b_frag(Br, ldb, k, nBase + 16 * j, lane);
            bi[j] = load_b_frag(Bi, ldb, k, nBase + 16 * j, lane);
        }
#pragma unroll
        for (int i = 0; i < 2; ++i)
#pragma unroll
            for (int j = 0; j < 2; ++j) {
                aRR[i][j] = __builtin_amdgcn_wmma_f32_16x16x4_f32(
                    false, ar[i], false, br[j], (short)0, aRR[i][j], false, false);
                aII[i][j] = __builtin_amdgcn_wmma_f32_16x16x4_f32(
                    false, ai[i], false, bi[j], (short)0, aII[i][j], false, false);
                aRI[i][j] = __builtin_amdgcn_wmma_f32_16x16x4_f32(
                    false, ar[i], false, bi[j], (short)0, aRI[i][j], false, false);
                aIR[i][j] = __builtin_amdgcn_wmma_f32_16x16x4_f32(
                    false, ai[i], false, br[j], (short)0, aIR[i][j], false, false);
            }
    }

#pragma unroll
    for (int i = 0; i < 2; ++i)
#pragma unroll
        for (int j = 0; j < 2; ++j) {
            v8f cr = aRR[i][j] - aII[i][j];     // Re = ArBr - AiBi
            v8f ci = aRI[i][j] + aIR[i][j];     // Im = ArBi + AiBr
            store_c_frag(Cr, ldc, mBase + 16 * i, nBase + 16 * j, lane, cr);
            store_c_frag(Ci, ldc, mBase + 16 * i, nBase + 16 * j, lane, ci);
        }
}

// ---------------- KB gather + centering phase -------------------------------
__global__ void gather_kernel(const float* __restrict__ Gr, const float* __restrict__ Gi,
                              const float* __restrict__ wy, const float* __restrict__ wx,
                              const float* __restrict__ phr, const float* __restrict__ phi,
                              const int* __restrict__ iy, const int* __restrict__ ix,
                              float* __restrict__ out, int N) {
    int n = blockIdx.x * blockDim.x + threadIdx.x;
    int b = blockIdx.y;
    if (n >= N) return;

    const float* gr = Gr + (size_t)b * GG * GG;
    const float* gi = Gi + (size_t)b * GG * GG;

    int   iyv[JJ];
    float wyv[JJ];
#pragma unroll
    for (int j = 0; j < JJ; ++j) {
        iyv[j] = iy[n * JJ + j] * GG;
        wyv[j] = wy[n * JJ + j];
    }

    float sr = 0.0f, si = 0.0f;
#pragma unroll
    for (int kx = 0; kx < JJ; ++kx) {
        int   xc = ix[n * JJ + kx];
        float wv = wx[n * JJ + kx];
        float rr = 0.0f, ri = 0.0f;
#pragma unroll
        for (int j = 0; j < JJ; ++j) {
            int lin = iyv[j] + xc;
            rr = fmaf(gr[lin], wyv[j], rr);
            ri = fmaf(gi[lin], wyv[j], ri);
        }
        sr = fmaf(rr, wv, sr);
        si = fmaf(ri, wv, si);
    }

    float pr = phr[n], pi = phi[n];
    size_t o = ((size_t)b * N + n) * 2;
    out[o + 0] = sr * pr - si * pi;
    out[o + 1] = sr * pi + si * pr;
}

// ---------------------------------------------------------------------------
extern "C" void kernel_launch(void* const* d_in, const int* in_sizes, int n_in,
                              void* d_out, int out_size, void* d_ws, size_t ws_size,
                              hipStream_t stream) {
    (void)n_in; (void)out_size; (void)ws_size;

    const float* xr   = (const float*)d_in[0];
    const float* xi   = (const float*)d_in[1];
    const float* sc_y = (const float*)d_in[2];
    const float* sc_x = (const float*)d_in[3];
    const float* wy   = (const float*)d_in[4];
    const float* wx   = (const float*)d_in[5];
    const float* phr  = (const float*)d_in[6];
    const float* phi  = (const float*)d_in[7];
    const int*   iy   = (const int*)d_in[8];
    const int*   ix   = (const int*)d_in[9];
    const int N = in_sizes[6];                 // phr has N elements

    // Workspace layout (floats), ~30 MB total, fully rewritten every call.
    float* ws = (float*)d_ws;
    float* Fc = ws;                            // 512*512
    float* Fs = Fc + GG * GG;                  // 512*512
    float* Xr = Fs + GG * GG;                  // B*256*256
    float* Xi = Xr + BB * HH * WW;
    float* Tr = Xi + BB * HH * WW;             // B*256*512
    float* Ti = Tr + BB * HH * GG;
    float* Gr = Ti + BB * HH * GG;             // B*512*512
    float* Gi = Gr + BB * GG * GG;

    // 1) DFT matrix (scaled 1/sqrt(512) so two passes give the 1/512 ortho norm)
    fgen_kernel<<<(GG * GG + 255) / 256, 256, 0, stream>>>(Fc, Fs);

    // 2) apodization
    apod_kernel<<<(BB * HH * WW + 255) / 256, 256, 0, stream>>>(xr, xi, sc_y, sc_x, Xr, Xi);

    // 3) pass 1: T (256x512) = X (256x256) * F[0:256, :]   (K=256 due to zero-pad)
    cgemm_wmma_kernel<<<dim3(HH / 64, GG / 128, BB), 256, 0, stream>>>(
        Xr, Xi, WW,  (long)HH * WW,
        Fc, Fs, GG,  0L,
        Tr, Ti, GG,  (long)HH * GG, WW);

    // 4) pass 2: Ghat (512x512) = F[:, 0:256] * T          (K=256)
    cgemm_wmma_kernel<<<dim3(GG / 64, GG / 128, BB), 256, 0, stream>>>(
        Fc, Fs, GG,  0L,
        Tr, Ti, GG,  (long)HH * GG,
        Gr, Gi, GG,  (long)GG * GG, HH);

    // 5) KB 7x7 gather + phase (grids are L2-resident: 16 MB << 192 MB L2)
    gather_kernel<<<dim3((N + 255) / 256, BB), 256, 0, stream>>>(
        Gr, Gi, wy, wx, phr, phi, iy, ix, (float*)d_out, N);
}